// VideoMAE_mask_one_video_64811056497333
// MI455X (gfx1250) — compile-verified
//
#include <hip/hip_runtime.h>
#include <hip/hip_bf16.h>

// ---------------------------------------------------------------------------
// VideoMAE patch-embed + mask for MI455X (gfx1250, wave32, WMMA).
// Only the 1248 visible tubelets (16 temporal x 78 kept spatial) are gathered
// and multiplied: D[1248x768] = A[1248x1536] @ W^T + b, fp32 via
// v_wmma_f32_16x16x4_f32. Gather uses GLOBAL_LOAD_ASYNC_TO_LDS_B128 (ASYNCcnt).
// Block = 32 M rows x full N=768 -> each B fragment feeds 2 WMMAs.
// ---------------------------------------------------------------------------

typedef float v2f __attribute__((ext_vector_type(2)));
typedef float v8f __attribute__((ext_vector_type(8)));

#define C_CH   3
#define H_PX   448
#define W_PX   448
#define TP     2
#define SP     16
#define EMB    768
#define KDIM   1536          // C*TP*SP*SP
#define TT     16            // T/TP
#define SPW    28            // W/SP (and H/SP)
#define SS     784           // 28*28
#define NKEEP  78
#define MROWS  (TT * NKEEP)  // 1248
#define MTILE  32            // rows per block (two 16-row WMMA tiles)
#define LDSROW 1540          // padded row stride (dwords) -> conflict-free frags
#define LDS_BYTES (MTILE * LDSROW * 4)   // 197120 B, fits 320KB/WGP

// ---------------------------------------------------------------------------
// GEMM kernel: one block = 32 M rows x full N (768), 8 waves.
// Wave w covers N columns [w*96, w*96+96) as six 16x16 tiles, for BOTH
// 16-row M sub-tiles -> 12 accumulators, 12 WMMA per 6 B-loads per K-step.
// ---------------------------------------------------------------------------
__global__ __launch_bounds__(256)
void vmae_visible_gemm(const float* __restrict__ clip,
                       const float* __restrict__ Wmat,   // (768, 1536) row-major
                       const float* __restrict__ bias,   // (768,)
                       const int*   __restrict__ keep,   // (78,) sorted
                       float*       __restrict__ out)    // (1248, 768)
{
    extern __shared__ float ldsmem[];   // MTILE rows x LDSROW floats, dyn-LDS
                                        // starts at LDS offset 0 (no static LDS)

    const int tid = threadIdx.x;
    const int m0  = blockIdx.x * MTILE;           // first visible-token row

    // ---------------- gather phase: 32 tubelet rows -> LDS (async) --------
    // Row m: t = m/78, keep-slot s = m%78, spatial patch p = keep[s].
    // k = ((c*2+tp)*16+sh)*16+sw ; 4 consecutive sw are contiguous in clip
    // memory -> 16B transfers. 32 rows * 384 chunks = 12288; 48 per thread.
    for (int i = tid; i < MTILE * 384; i += 256) {
        const int row = i / 384;         // 0..31
        const int c4  = i % 384;         // float4 chunk within row
        const int m   = m0 + row;
        const int t   = m / NKEEP;
        const int s   = m % NKEEP;
        const int p   = keep[s];
        const int ph  = p / SPW;
        const int pw  = p % SPW;

        const int sw  = (c4 & 3) * 4;
        const int sh  = (c4 >> 2) & 15;
        const int tp  = (c4 >> 6) & 1;
        const int c   = c4 >> 7;

        const long off = (((long)(t * TP + tp) * C_CH + c) * H_PX
                          + (ph * SP + sh)) * W_PX + (pw * SP + sw);
        const float*   gsrc   = clip + off;
        const unsigned ldsOff = (unsigned)((row * LDSROW + c4 * 4) * 4);
        // Memory -> LDS directly, tracked by ASYNCcnt (no VGPR round-trip).
        asm volatile("global_load_async_to_lds_b128 %0, %1, off"
                     :: "v"(ldsOff), "v"(gsrc) : "memory");
    }
    asm volatile("s_wait_asynccnt 0" ::: "memory");
    __syncthreads();

    // ---------------- WMMA phase ------------------------------------------
    const int wave  = tid >> 5;          // 0..7
    const int lane  = tid & 31;
    const int lrow  = lane & 15;         // A row / B,C,D column-within-tile
    const int khalf = (lane >> 4) * 2;   // lanes 16-31 hold K+2,K+3
    const int nb    = wave * 96;         // wave's first N column

    v8f acc[2][6];
    #pragma unroll
    for (int mt = 0; mt < 2; ++mt)
        #pragma unroll
        for (int j = 0; j < 6; ++j) acc[mt][j] = (v8f)0.0f;

    const float* aRow0 = ldsmem + lrow * LDSROW + khalf;          // M rows 0-15
    const float* aRow1 = ldsmem + (16 + lrow) * LDSROW + khalf;   // M rows 16-31

    // per-tile B row pointers: lane reads W[n][k0+khalf .. +1]
    const float* bPtr[6];
    #pragma unroll
    for (int j = 0; j < 6; ++j)
        bPtr[j] = Wmat + (long)(nb + j * 16 + lrow) * KDIM + khalf;

    #pragma unroll 2
    for (int k0 = 0; k0 < KDIM; k0 += 4) {
        // A fragments: VGPR0 = A[row][k0+khalf], VGPR1 = A[row][k0+khalf+1]
        const v2f a0 = *(const v2f*)(aRow0 + k0);
        const v2f a1 = *(const v2f*)(aRow1 + k0);
        __builtin_prefetch(bPtr[0] + k0 + 64, 0, 0);   // pull next W chunk
        #pragma unroll
        for (int j = 0; j < 6; ++j) {
            const v2f bf = *(const v2f*)(bPtr[j] + k0);
            acc[0][j] = __builtin_amdgcn_wmma_f32_16x16x4_f32(
                false, a0, false, bf, (short)0, acc[0][j], false, false);
            acc[1][j] = __builtin_amdgcn_wmma_f32_16x16x4_f32(
                false, a1, false, bf, (short)0, acc[1][j], false, false);
        }
    }

    // ---------------- bias + store ----------------------------------------
    // C/D layout: VGPR v, lanes 0-15 -> (M=v, N=lane); lanes 16-31 -> (M=v+8).
    const int moff = (lane >> 4) * 8;
    #pragma unroll
    for (int mt = 0; mt < 2; ++mt) {
        #pragma unroll
        for (int j = 0; j < 6; ++j) {
            const int   n  = nb + j * 16 + lrow;
            const float bn = bias[n];
            #pragma unroll
            for (int v = 0; v < 8; ++v) {
                const int m = m0 + mt * 16 + v + moff;
                out[(long)m * EMB + n] = acc[mt][j][v] + bn;
            }
        }
    }
}

// ---------------------------------------------------------------------------
// Mask kernel: mask_full (784,16) written as 0.0/1.0 after the token region.
// ---------------------------------------------------------------------------
__global__ __launch_bounds__(256)
void vmae_mask(const int* __restrict__ keep, float* __restrict__ maskOut)
{
    const int i = blockIdx.x * blockDim.x + threadIdx.x;
    if (i >= SS * TT) return;
    const int ss = i / TT;
    // binary search sorted keep[0..77] for ss
    int lo = 0, hi = NKEEP - 1, found = 0;
    while (lo <= hi) {
        const int mid = (lo + hi) >> 1;
        const int v = keep[mid];
        if (v == ss) { found = 1; break; }
        if (v < ss) lo = mid + 1; else hi = mid - 1;
    }
    maskOut[i] = found ? 1.0f : 0.0f;
}

// ---------------------------------------------------------------------------
extern "C" void kernel_launch(void* const* d_in, const int* in_sizes, int n_in,
                              void* d_out, int out_size, void* d_ws, size_t ws_size,
                              hipStream_t stream) {
    const float* clip = (const float*)d_in[0];   // (1,32,3,448,448)
    const float* Wm   = (const float*)d_in[1];   // (768,1536)
    const float* bias = (const float*)d_in[2];   // (768,)
    const int*   keep = (const int*)  d_in[3];   // (78,)

    float* tokens = (float*)d_out;               // (1248,768)
    float* mask   = tokens + (long)MROWS * EMB;  // (784,16) as float

    vmae_visible_gemm<<<MROWS / MTILE, 256, LDS_BYTES, stream>>>(
        clip, Wm, bias, keep, tokens);

    vmae_mask<<<(SS * TT + 255) / 256, 256, 0, stream>>>(keep, mask);
}